// CloudMatrixModel2_9474697855432
// MI455X (gfx1250) — compile-verified
//
#include <hip/hip_runtime.h>
#include <math.h>

// ---------------------------------------------------------------------------
// CloudMatrixModel2 (MatNet mixed-score encoder) for MI455X / gfx1250.
// fp32 end-to-end; GEMMs via V_WMMA_F32_16X16X4_F32 (wave32 matrix pipe).
// Workload is launch/bandwidth-bound (25 GFLOP, ~21 MB of weights, all
// activations L2-resident), so exact fp32 matrix ops are the right precision.
// ---------------------------------------------------------------------------

typedef float v2f __attribute__((ext_vector_type(2)));
typedef float v8f __attribute__((ext_vector_type(8)));

#define EDIM 256
#define HN   16
#define DD   16
#define MSD  16
#define FFD  512
#define LN   5
#define BB   8
#define TT   256
#define MM   128

// ---------------------------------------------------------------------------
// Batched GEMM: C[b] = act(A[b] @ W(+batched/transposed) + bias)
// grid = (N/64, rows/16, B), block = 128 (4 waves; wave w -> 16-col slab).
// One wave owns one 16x16 output tile, K marched in steps of 4 through WMMA.
// BT/ACT are template params so the inner loop is branch-free.
// WMMA f32 16x16x4 layouts (ISA 7.12.2):
//   A: lane L -> row L%16, vgpr j = A[row][k0 + 2*(L/16) + j]
//   B: lane L -> col L%16, vgpr j = W[k0 + 2*(L/16) + j][col]
//   D: lane L -> col L%16, vgpr j = D[j + 8*(L/16)][col]
// ---------------------------------------------------------------------------
template <bool BT, bool ACT>
__global__ __launch_bounds__(128)
void gemm_f32_wmma(const float* __restrict__ A, long sA,
                   const float* __restrict__ Bm, long sB,
                   const float* __restrict__ bias,
                   float* __restrict__ C, long sC,
                   int K, int N)
{
  const int lane = threadIdx.x & 31;
  const int wave = threadIdx.x >> 5;
  const int half = lane >> 4;   // 0: K pair {0,1}, 1: K pair {2,3}
  const int lm   = lane & 15;
  const int row0 = blockIdx.y << 4;
  const int col  = (blockIdx.x << 6) + (wave << 4) + lm;
  const int r    = row0 + lm;

  // Per-lane base pointers (K is a multiple of 8, kk even -> 8B aligned).
  const float* Ap = A + (long)blockIdx.z * sA + (long)r * K + (half << 1);
  const float* Bp;
  if (BT) Bp = Bm + (long)blockIdx.z * sB + (long)col * K + (half << 1);
  else    Bp = Bm + (long)blockIdx.z * sB + (long)(half << 1) * N + col;

  v8f acc = {};
#pragma unroll 2
  for (int k0 = 0; k0 < K; k0 += 4) {
    v2f a = *(const v2f*)(Ap);
    v2f b;
    if (BT) {
      b = *(const v2f*)(Bp);
      Bp += 4;
    } else {
      b.x = Bp[0];
      b.y = Bp[N];
      Bp += (long)4 * N;
    }
    Ap += 4;
    acc = __builtin_amdgcn_wmma_f32_16x16x4_f32(
        /*neg_a=*/false, a, /*neg_b=*/false, b,
        /*c_mod=*/(short)0, acc, /*reuse_a=*/false, /*reuse_b=*/false);
  }

  const float bv = bias ? bias[col] : 0.0f;
  float* Cb = C + (long)blockIdx.z * sC;
#pragma unroll
  for (int j = 0; j < 8; ++j) {
    float x = acc[j] + bv;
    if (ACT) x = fmaxf(x, 0.0f);
    Cb[(long)(row0 + (half << 3) + j) * N + col] = x;
  }
}

// ---------------------------------------------------------------------------
// Mixed-score attention for one (batch, head). K/V head slices staged in LDS.
// One thread per query row; fused 2->16->1 MLP on (score, cost) + online
// softmax + weighted V accumulation (flash style, single pass over m).
// use_mlp==0 -> plain scaled-dot attention (decoder).
// ---------------------------------------------------------------------------
__global__ __launch_bounds__(256)
void attn_mixed(const float* __restrict__ q, const float* __restrict__ k,
                const float* __restrict__ v,
                const float* __restrict__ cost, int cSN, int cSM,
                const float* __restrict__ m1w, const float* __restrict__ m1b,
                const float* __restrict__ m2w, const float* __restrict__ m2b,
                float* __restrict__ out, int n, int m, int use_mlp)
{
  __shared__ float Ks[TT * DD];
  __shared__ float Vs[TT * DD];
  const int h = blockIdx.x;
  const int b = blockIdx.y;
  const int tid = threadIdx.x;

  for (int idx = tid; idx < m * DD; idx += 256) {
    const int rr = idx >> 4, d = idx & 15;
    Ks[idx] = k[((long)(b * m + rr)) * EDIM + h * DD + d];
    Vs[idx] = v[((long)(b * m + rr)) * EDIM + h * DD + d];
  }
  __syncthreads();

  float w0[MSD], w1[MSD], bb1[MSD], w2[MSD], b2 = 0.0f;
  if (use_mlp) {
    const float* m1wh = m1w + (h * 2) * MSD;   // (H,2,MS)
#pragma unroll
    for (int o = 0; o < MSD; ++o) {
      w0[o]  = m1wh[o];
      w1[o]  = m1wh[MSD + o];
      bb1[o] = m1b[h * MSD + o];
      w2[o]  = m2w[h * MSD + o];
    }
    b2 = m2b[h];
  }
  const float* costb = cost + (long)b * TT * MM;

  for (int r = tid; r < n; r += 256) {
    float qr[DD];
#pragma unroll
    for (int d = 0; d < DD; ++d)
      qr[d] = q[((long)(b * n + r)) * EDIM + h * DD + d];

    float mx = -3.0e38f, sm = 0.0f;
    float acc[DD];
#pragma unroll
    for (int d = 0; d < DD; ++d) acc[d] = 0.0f;

    for (int mm = 0; mm < m; ++mm) {
      float s = 0.0f;
#pragma unroll
      for (int d = 0; d < DD; ++d) s += qr[d] * Ks[mm * DD + d];
      s *= 0.25f;                        // / SQRT_D
      if (use_mlp) {
        const float cst = costb[(long)r * cSN + (long)mm * cSM];
        float t = b2;
#pragma unroll
        for (int o = 0; o < MSD; ++o)
          t += fmaxf(fmaf(s, w0[o], fmaf(cst, w1[o], bb1[o])), 0.0f) * w2[o];
        s = t;
      }
      const float nmx = fmaxf(mx, s);
      const float e = __expf(s - nmx);
      const float f = __expf(mx - nmx);
      sm = fmaf(sm, f, e);
#pragma unroll
      for (int d = 0; d < DD; ++d)
        acc[d] = fmaf(acc[d], f, e * Vs[mm * DD + d]);
      mx = nmx;
    }
    const float inv = 1.0f / sm;
#pragma unroll
    for (int d = 0; d < DD; ++d)
      out[((long)(b * n + r)) * EDIM + h * DD + d] = acc[d] * inv;
  }
}

// ---------------------------------------------------------------------------
// out[b, :, c] = inorm(X[b,:,c] + Y[b,:,c]) over the sequence dim.
// grid = (B), block = 256 (one thread per channel -> fully coalesced rows;
// each thread privately reduces its own channel, no LDS needed).
// ---------------------------------------------------------------------------
__global__ __launch_bounds__(256)
void add_inorm(const float* __restrict__ X, const float* __restrict__ Y,
               const float* __restrict__ scale, const float* __restrict__ bias,
               float* __restrict__ out, int n)
{
  const int b = blockIdx.x;
  const int c = threadIdx.x;
  const long base = (long)b * n * EDIM + c;

  float s = 0.0f, sq = 0.0f;
  for (int r = 0; r < n; ++r) {
    const float x = X[base + (long)r * EDIM] + Y[base + (long)r * EDIM];
    s += x;
    sq = fmaf(x, x, sq);
  }
  const float mu  = s / (float)n;
  const float var = sq / (float)n - mu * mu;
  const float inv = rsqrtf(var + 1e-5f) * scale[c];
  const float bs  = bias[c];
  for (int r = 0; r < n; ++r) {
    const float x = X[base + (long)r * EDIM] + Y[base + (long)r * EDIM];
    out[base + (long)r * EDIM] = (x - mu) * inv + bs;
  }
}

// ---------------------------------------------------------------------------
// logits = 10*tanh(score/16) + mask; softmax over flattened M*T per batch.
// ---------------------------------------------------------------------------
__global__ __launch_bounds__(256)
void final_softmax(const float* __restrict__ score, const float* __restrict__ mask,
                   float* __restrict__ out, int len)
{
  __shared__ float red[256];
  const int b = blockIdx.x;
  const int tid = threadIdx.x;
  const float* sb = score + (long)b * len;
  const float* kb = mask  + (long)b * len;

  float mx = -3.0e38f;
  for (int i = tid; i < len; i += 256)
    mx = fmaxf(mx, 10.0f * tanhf(sb[i] * 0.0625f) + kb[i]);
  red[tid] = mx; __syncthreads();
  for (int off = 128; off > 0; off >>= 1) {
    if (tid < off) red[tid] = fmaxf(red[tid], red[tid + off]);
    __syncthreads();
  }
  mx = red[0]; __syncthreads();

  float sm = 0.0f;
  for (int i = tid; i < len; i += 256)
    sm += __expf(10.0f * tanhf(sb[i] * 0.0625f) + kb[i] - mx);
  red[tid] = sm; __syncthreads();
  for (int off = 128; off > 0; off >>= 1) {
    if (tid < off) red[tid] += red[tid + off];
    __syncthreads();
  }
  const float inv = 1.0f / red[0];

  float* ob = out + (long)b * len;
  for (int i = tid; i < len; i += 256)
    ob[i] = __expf(10.0f * tanhf(sb[i] * 0.0625f) + kb[i] - mx) * inv;
}

// ---------------------------------------------------------------------------
extern "C" void kernel_launch(void* const* d_in, const int* in_sizes, int n_in,
                              void* d_out, int out_size, void* d_ws, size_t ws_size,
                              hipStream_t stream)
{
  const float* machine_state = (const float*)d_in[0];
  const float* task_state    = (const float*)d_in[1];
  const float* D_TM          = (const float*)d_in[2];
  const float* ninf          = (const float*)d_in[3];
  const float* T_W  = (const float*)d_in[4];
  const float* T_b  = (const float*)d_in[5];
  const float* M_W  = (const float*)d_in[6];
  const float* M_b  = (const float*)d_in[7];
  const float* eWq  = (const float*)d_in[8];
  const float* eWk  = (const float*)d_in[9];
  const float* eWv  = (const float*)d_in[10];
  const float* em1w = (const float*)d_in[11];
  const float* em1b = (const float*)d_in[12];
  const float* em2w = (const float*)d_in[13];
  const float* em2b = (const float*)d_in[14];
  const float* ecW  = (const float*)d_in[15];
  const float* ecb  = (const float*)d_in[16];
  const float* en1s = (const float*)d_in[17];
  const float* en1b = (const float*)d_in[18];
  const float* efW1 = (const float*)d_in[19];
  const float* efb1 = (const float*)d_in[20];
  const float* efW2 = (const float*)d_in[21];
  const float* efb2 = (const float*)d_in[22];
  const float* en2s = (const float*)d_in[23];
  const float* en2b = (const float*)d_in[24];
  const float* dWq  = (const float*)d_in[25];
  const float* dWk  = (const float*)d_in[26];
  const float* dWv  = (const float*)d_in[27];
  const float* dcW  = (const float*)d_in[28];
  const float* dcb  = (const float*)d_in[29];

  // Workspace layout (floats); total ~23 MB.
  float* ws     = (float*)d_ws;
  float* rowA   = ws;                        // B*T*E
  float* rowB   = rowA   + (long)BB*TT*EDIM; // B*T*E
  float* colbuf = rowB   + (long)BB*TT*EDIM; // B*M*E
  float* qb     = colbuf + (long)BB*MM*EDIM; // B*T*E
  float* kb     = qb     + (long)BB*TT*EDIM; // B*T*E
  float* vb     = kb     + (long)BB*TT*EDIM; // B*T*E
  float* catb   = vb     + (long)BB*TT*EDIM; // B*T*E
  float* tmp1   = catb   + (long)BB*TT*EDIM; // B*T*E
  float* o1b    = tmp1   + (long)BB*TT*EDIM; // B*T*E
  float* ffb    = o1b    + (long)BB*TT*EDIM; // B*T*FF
  float* scoreb = ffb    + (long)BB*TT*FFD;  // B*M*T

  auto gemm = [&](const float* A, long sA, const float* Bm, long sB,
                  const float* bias, float* C, int rows, int K, int N,
                  int act, int bT) {
    dim3 g((unsigned)(N / 64), (unsigned)(rows / 16), (unsigned)BB);
    const long sC = (long)rows * N;
    if (bT)
      gemm_f32_wmma<true, false><<<g, dim3(128), 0, stream>>>(A, sA, Bm, sB, bias, C, sC, K, N);
    else if (act)
      gemm_f32_wmma<false, true><<<g, dim3(128), 0, stream>>>(A, sA, Bm, sB, bias, C, sC, K, N);
    else
      gemm_f32_wmma<false, false><<<g, dim3(128), 0, stream>>>(A, sA, Bm, sB, bias, C, sC, K, N);
  };

  // ---- embeddings ----
  gemm(task_state,    (long)TT * 16, T_W, 0, T_b, rowA,   TT, 16, EDIM, 1, 0);
  gemm(machine_state, (long)MM * 8,  M_W, 0, M_b, colbuf, MM, 8,  EDIM, 1, 0);

  // ---- encoder layers ----
  float* row  = rowA;
  float* nrow = rowB;
  for (int l = 0; l < LN; ++l) {
    for (int s = 0; s < 2; ++s) {
      const long wo = (long)(l * 2 + s);
      const float* Asrc = (s == 0) ? row : colbuf;      // queries + residual
      const float* Csrc = (s == 0) ? colbuf : row;      // keys/values (old row!)
      const int n = (s == 0) ? TT : MM;
      const int m = (s == 0) ? MM : TT;
      float* dst = (s == 0) ? nrow : colbuf;
      const int cSN = (s == 0) ? MM : 1;                // cost strides: D_TM / D_TM^T
      const int cSM = (s == 0) ? 1  : MM;

      gemm(Asrc, (long)n * EDIM, eWq + wo * EDIM * 256, 0, nullptr, qb, n, EDIM, 256, 0, 0);
      gemm(Csrc, (long)m * EDIM, eWk + wo * EDIM * 256, 0, nullptr, kb, m, EDIM, 256, 0, 0);
      gemm(Csrc, (long)m * EDIM, eWv + wo * EDIM * 256, 0, nullptr, vb, m, EDIM, 256, 0, 0);

      attn_mixed<<<dim3(HN, BB), dim3(256), 0, stream>>>(
          qb, kb, vb, D_TM, cSN, cSM,
          em1w + wo * HN * 2 * MSD, em1b + wo * HN * MSD,
          em2w + wo * HN * MSD,     em2b + wo * HN,
          catb, n, m, 1);

      gemm(catb, (long)n * EDIM, ecW + wo * 256 * EDIM, 0, ecb + wo * EDIM,
           tmp1, n, 256, EDIM, 0, 0);
      add_inorm<<<dim3(BB), dim3(EDIM), 0, stream>>>(
          Asrc, tmp1, en1s + wo * EDIM, en1b + wo * EDIM, o1b, n);

      gemm(o1b, (long)n * EDIM, efW1 + wo * EDIM * FFD, 0, efb1 + wo * FFD,
           ffb, n, EDIM, FFD, 1, 0);
      gemm(ffb, (long)n * FFD,  efW2 + wo * FFD * EDIM, 0, efb2 + wo * EDIM,
           tmp1, n, FFD, EDIM, 0, 0);
      add_inorm<<<dim3(BB), dim3(EDIM), 0, stream>>>(
          o1b, tmp1, en2s + wo * EDIM, en2b + wo * EDIM, dst, n);
    }
    float* t = row; row = nrow; nrow = t;
  }

  // ---- decoder: machines query tasks ----
  gemm(colbuf, (long)MM * EDIM, dWq, 0, nullptr, qb, MM, EDIM, 256, 0, 0);
  gemm(row,    (long)TT * EDIM, dWk, 0, nullptr, kb, TT, EDIM, 256, 0, 0);
  gemm(row,    (long)TT * EDIM, dWv, 0, nullptr, vb, TT, EDIM, 256, 0, 0);

  attn_mixed<<<dim3(HN, BB), dim3(256), 0, stream>>>(
      qb, kb, vb, D_TM, 0, 0, nullptr, nullptr, nullptr, nullptr,
      catb, MM, TT, 0);

  gemm(catb, (long)MM * EDIM, dcW, 0, dcb, tmp1, MM, 256, EDIM, 0, 0);

  // score[b] = mh[b] @ row[b]^T   (batched B, transposed)
  gemm(tmp1, (long)MM * EDIM, row, (long)TT * EDIM, nullptr, scoreb,
       MM, EDIM, TT, 0, 1);

  final_softmax<<<dim3(BB), dim3(256), 0, stream>>>(
      scoreb, ninf, (float*)d_out, MM * TT);
}